// QuantumEntanglingLinearVectorized_43215960932475
// MI455X (gfx1250) — compile-verified
//
#include <hip/hip_runtime.h>
#include <hip/hip_bf16.h>

// Givens-rotation layers, DIM=4096, 4 layers x 3 stages. Memory-bound:
// 256 MB HBM traffic -> ~11us floor @ 23.3 TB/s. Strategy: async global->LDS
// (ASYNCcnt path), one 16-elem chunk per thread held in REGISTERS across all
// 12 stages (all pairs are adjacent), odd stages exchange 2 boundary floats
// per thread per layer via LDS, then async LDS->global writeback.

#define DIM     4096
#define PAIRS   2048            // DIM/2
#define NLAYERS 4
#define NSTAGES (NLAYERS * 3)
#define BLOCK   256             // 8 waves (wave32); one row per block
#define CHUNK   16              // elements per thread (8 pairs)
#define CH4     4               // float4s per chunk
#define PADF4   5               // padded float4 slots per chunk (20 floats)
                                //   -> 32-lane b128 hits every bank exactly 2x

__device__ __forceinline__ void rot1(float& a, float& b, float c, float s) {
    float xi = a, xj = b;
    a = fmaf(c, xi,  s * xj);   //  c*xi + s*xj
    b = fmaf(c, xj, -s * xi);   // -s*xi + c*xj
}

// ---------------------------------------------------------------------------
// Kernel 1: cos/sin tables. trig[sg][0..2047]=cos, trig[sg][2048..4095]=sin.
// grid = (PAIRS/BLOCK, NSTAGES)
// ---------------------------------------------------------------------------
__global__ __launch_bounds__(BLOCK)
void qe_trig_kernel(const float* __restrict__ local_angles,
                    const float* __restrict__ ent_angles,
                    float* __restrict__ trig)
{
    const int k     = blockIdx.x * BLOCK + threadIdx.x;   // 0..2047
    const int sg    = blockIdx.y;                         // 0..11
    const int l     = sg / 3;
    const int stage = sg - l * 3;

    float theta;
    if (stage == 0)      theta = local_angles[l * DIM + 2 * k];      // even pairs
    else if (stage == 1) theta = local_angles[l * DIM + 2 * k + 1];  // odd pairs
    else                 theta = ent_angles[l * PAIRS + k];          // even pairs

    float s, c;
    sincosf(theta, &s, &c);
    trig[sg * (2 * PAIRS) + k]         = c;
    trig[sg * (2 * PAIRS) + PAIRS + k] = s;
}

// ---------------------------------------------------------------------------
// Kernel 2: one row per block; 16 contiguous elements per thread in registers.
// ---------------------------------------------------------------------------
__global__ __launch_bounds__(BLOCK)
void qe_rot_kernel(const float* __restrict__ x,
                   const float* __restrict__ trig,
                   float* __restrict__ out)
{
    __shared__ float4 tile4[BLOCK * PADF4];   // 20 KB padded row staging
    __shared__ float  xf[2][BLOCK];           // chunk-first exchange (dbl-buf)
    __shared__ float  xl[2][BLOCK];           // chunk-last  exchange (dbl-buf)

    const int t = threadIdx.x;
    const long long rowBase = (long long)blockIdx.x * DIM;
    // Low 32 bits of generic pointer = wave-relative LDS byte offset.
    const unsigned lds_base = (unsigned)(uintptr_t)(void*)&tile4[0];

    // ---- async load: row float4 j -> padded slot (j/4)*PADF4 + (j%4) ----
    #pragma unroll
    for (int i = 0; i < (DIM / 4) / BLOCK; ++i) {         // 4 issues/thread
        int j = i * BLOCK + t;
        unsigned lds_off = lds_base + (unsigned)(((j >> 2) * PADF4 + (j & 3)) * 16);
        unsigned long long gaddr =
            (unsigned long long)(uintptr_t)(x + rowBase + (long long)j * 4);
        asm volatile("global_load_async_to_lds_b128 %0, %1, off"
                     :: "v"(lds_off), "v"(gaddr) : "memory");
    }
    asm volatile("s_wait_asynccnt 0" ::: "memory");
    __syncthreads();

    // ---- padded LDS -> registers (bank-optimal b128 reads) ----
    float v[CHUNK];
    {
        const float4* base = &tile4[t * PADF4];
        #pragma unroll
        for (int q = 0; q < CH4; ++q) {
            float4 d = base[q];
            v[4*q+0] = d.x; v[4*q+1] = d.y; v[4*q+2] = d.z; v[4*q+3] = d.w;
        }
    }

    const float4* __restrict__ trig4 = (const float4*)trig;

    #pragma unroll 1
    for (int l = 0; l < NLAYERS; ++l) {
        const int sgA = 3 * l, sgB = 3 * l + 1, sgC = 3 * l + 2;

        // ---- stage A: even pairs (v[2m], v[2m+1]), k = 8t+m ----
        {
            float4 c0 = trig4[sgA * 1024 + 2 * t];
            float4 c1 = trig4[sgA * 1024 + 2 * t + 1];
            float4 s0 = trig4[sgA * 1024 + 512 + 2 * t];
            float4 s1 = trig4[sgA * 1024 + 512 + 2 * t + 1];
            float c[8] = {c0.x, c0.y, c0.z, c0.w, c1.x, c1.y, c1.z, c1.w};
            float s[8] = {s0.x, s0.y, s0.z, s0.w, s1.x, s1.y, s1.z, s1.w};
            #pragma unroll
            for (int m = 0; m < 8; ++m) rot1(v[2*m], v[2*m+1], c[m], s[m]);
        }

        // ---- stage B: odd pairs; boundary floats exchanged via LDS ----
        xf[l & 1][t] = v[0];
        xl[l & 1][t] = v[CHUNK - 1];
        __syncthreads();
        {
            float nf = xf[l & 1][(t + 1) & (BLOCK - 1)];          // next chunk first
            float pl = xl[l & 1][(t - 1) & (BLOCK - 1)];          // prev chunk last

            float4 c0 = trig4[sgB * 1024 + 2 * t];
            float4 c1 = trig4[sgB * 1024 + 2 * t + 1];
            float4 s0 = trig4[sgB * 1024 + 512 + 2 * t];
            float4 s1 = trig4[sgB * 1024 + 512 + 2 * t + 1];
            float c[8] = {c0.x, c0.y, c0.z, c0.w, c1.x, c1.y, c1.z, c1.w};
            float s[8] = {s0.x, s0.y, s0.z, s0.w, s1.x, s1.y, s1.z, s1.w};

            const float* ct = trig + sgB * (2 * PAIRS);
            int kl = (8 * t + PAIRS - 1) & (PAIRS - 1);           // left pair idx
            float cl = ct[kl];
            float sl = ct[PAIRS + kl];

            // interior: (v[2m+1], v[2m+2]), k = 8t+m, m=0..6
            #pragma unroll
            for (int m = 0; m < 7; ++m) rot1(v[2*m+1], v[2*m+2], c[m], s[m]);
            // right boundary pair (elem 16t+15 is i-side), k = 8t+7
            v[CHUNK - 1] = fmaf(c[7], v[CHUNK - 1], s[7] * nf);
            // left boundary pair (elem 16t is j-side), k = (8t-1) mod 2048
            v[0] = fmaf(cl, v[0], -sl * pl);
        }

        // ---- stage C: even pairs with ent angles ----
        {
            float4 c0 = trig4[sgC * 1024 + 2 * t];
            float4 c1 = trig4[sgC * 1024 + 2 * t + 1];
            float4 s0 = trig4[sgC * 1024 + 512 + 2 * t];
            float4 s1 = trig4[sgC * 1024 + 512 + 2 * t + 1];
            float c[8] = {c0.x, c0.y, c0.z, c0.w, c1.x, c1.y, c1.z, c1.w};
            float s[8] = {s0.x, s0.y, s0.z, s0.w, s1.x, s1.y, s1.z, s1.w};
            #pragma unroll
            for (int m = 0; m < 8; ++m) rot1(v[2*m], v[2*m+1], c[m], s[m]);
        }
    }

    // ---- registers -> padded LDS ----
    {
        float4* base = &tile4[t * PADF4];
        #pragma unroll
        for (int q = 0; q < CH4; ++q)
            base[q] = make_float4(v[4*q+0], v[4*q+1], v[4*q+2], v[4*q+3]);
    }
    asm volatile("s_wait_dscnt 0" ::: "memory");
    __syncthreads();

    // ---- async store: padded LDS -> linear global ----
    #pragma unroll
    for (int i = 0; i < (DIM / 4) / BLOCK; ++i) {
        int j = i * BLOCK + t;
        unsigned lds_off = lds_base + (unsigned)(((j >> 2) * PADF4 + (j & 3)) * 16);
        unsigned long long gaddr =
            (unsigned long long)(uintptr_t)(out + rowBase + (long long)j * 4);
        asm volatile("global_store_async_from_lds_b128 %0, %1, off"
                     :: "v"(gaddr), "v"(lds_off) : "memory");
    }
    asm volatile("s_wait_asynccnt 0" ::: "memory");
}

// ---------------------------------------------------------------------------
extern "C" void kernel_launch(void* const* d_in, const int* in_sizes, int n_in,
                              void* d_out, int out_size, void* d_ws, size_t ws_size,
                              hipStream_t stream)
{
    const float* x            = (const float*)d_in[0];  // (8192, 4096) f32
    const float* local_angles = (const float*)d_in[1];  // (4, 4096) f32
    const float* ent_angles   = (const float*)d_in[2];  // (4, 2048) f32
    float*       out          = (float*)d_out;          // (8192, 4096) f32
    float*       trig         = (float*)d_ws;           // 12*2*2048 floats = 192 KB

    const int n_rows = in_sizes[0] / DIM;               // 8192

    dim3 tgrid(PAIRS / BLOCK, NSTAGES);                 // (8, 12)
    qe_trig_kernel<<<tgrid, BLOCK, 0, stream>>>(local_angles, ent_angles, trig);

    qe_rot_kernel<<<n_rows, BLOCK, 0, stream>>>(x, trig, out);
}